// SEAL_GCN_14370960573130
// MI455X (gfx1250) — compile-verified
//
#include <hip/hip_runtime.h>
#include <hip/hip_bf16.h>

typedef float v2f __attribute__((ext_vector_type(2)));
typedef float v8f __attribute__((ext_vector_type(8)));

#define NFEAT 128
#define NGRAPH 512
#define NREL 200

// ---------------------------------------------------------------- utilities
__global__ void zero_f32_kernel(float* __restrict__ p, long n) {
    long i = (long)blockIdx.x * blockDim.x + threadIdx.x;
    if (i < n) p[i] = 0.0f;
}

// ---------------------------------------------------------------- degrees
__global__ void degree_kernel(const int* __restrict__ src, const int* __restrict__ dst,
                              float* __restrict__ deg_out, float* __restrict__ deg_in, int E) {
    int e = blockIdx.x * blockDim.x + threadIdx.x;
    if (e < E) {
        atomicAdd(&deg_out[src[e]], 1.0f);
        atomicAdd(&deg_in[dst[e]], 1.0f);
    }
}

__global__ void norm_finalize_kernel(float* __restrict__ d, long n) {
    long i = (long)blockIdx.x * blockDim.x + threadIdx.x;
    if (i < n) d[i] = rsqrtf(fmaxf(d[i], 1.0f));
}

// ---------------------------------------------------------------- features
__global__ void build_x_kernel(const int* __restrict__ nid, const int* __restrict__ z,
                               const float* __restrict__ emb, const float* __restrict__ ztab,
                               float* __restrict__ x, int N) {
    long t = (long)blockIdx.x * blockDim.x + threadIdx.x;  // t = i*128 + f
    long i = t >> 7;
    int  f = (int)(t & 127);
    if (i < N) {
        x[t] = (f < 64) ? emb[(long)nid[i] * 64 + f]
                        : ztab[(long)z[i] * 64 + (f - 64)];
    }
}

// ---------------------------------------------------------------- SpMM scatter
// agg[dst, :] += x[src, :] * norm_out[src]   (32 threads/edge, float4 each)
__global__ void scatter_kernel(const int* __restrict__ src, const int* __restrict__ dst,
                               const float* __restrict__ x, const float* __restrict__ norm_out,
                               float* __restrict__ agg, int E) {
    long t = (long)blockIdx.x * blockDim.x + threadIdx.x;
    long e = t >> 5;
    if (e >= E) return;
    int f4 = (int)((t & 31) << 2);
    int s = src[e], d = dst[e];
    float no = norm_out[s];
    const float4 v = *(const float4*)(x + (long)s * NFEAT + f4);
    float* ap = agg + (long)d * NFEAT + f4;
    atomicAdd(ap + 0, v.x * no);
    atomicAdd(ap + 1, v.y * no);
    atomicAdd(ap + 2, v.z * no);
    atomicAdd(ap + 3, v.w * no);
}

// ---------------------------------------------------------------- WMMA GEMM
// Y[nrows,128] = relu?( (A * norm[row]) @ W[128,128] + bias[128] )
// One wave -> 16 rows x 128 cols (8 tiles of 16x16), K looped in steps of 4
// with V_WMMA_F32_16X16X4_F32. W staged in LDS (64 KB).
__global__ __launch_bounds__(256) void gemm128_wmma_kernel(
    const float* __restrict__ A, const float* __restrict__ norm,
    const float* __restrict__ W, const float* __restrict__ bias,
    float* __restrict__ Y, int nrows, int relu) {
    __shared__ float Wlds[NFEAT * NFEAT];  // 64 KB

    int tid = threadIdx.x;
    {   // cooperative load of W into LDS: 4096 float4 over 256 threads
        const float4* Wv = (const float4*)W;
        float4* Lv = (float4*)Wlds;
#pragma unroll
        for (int i = 0; i < 16; ++i) Lv[tid + i * 256] = Wv[tid + i * 256];
    }
    __syncthreads();

    int wave = tid >> 5;               // wave-uniform
    int lane = tid & 31;
    int rbase = blockIdx.x * 128 + wave * 16;
    if (rbase >= nrows) return;        // uniform per wave -> EXEC stays all-ones

    int lrow  = lane & 15;             // M (for A) / N (for B,C,D) within tile
    int khalf = (lane >> 4) << 1;      // 0 for lanes 0-15, 2 for lanes 16-31
    int rowhi = (lane >> 4) << 3;      // C/D: lanes 16-31 hold M = r + 8

    long m = rbase + lrow;
    float nrm = norm ? norm[m] : 1.0f;
    const float* Arow = A + m * NFEAT;

    v8f acc[8];
#pragma unroll
    for (int t = 0; t < 8; ++t) {
        float bv = bias[t * 16 + lrow];   // C element depends only on column
#pragma unroll
        for (int r = 0; r < 8; ++r) acc[t][r] = bv;
    }

    for (int k = 0; k < NFEAT; k += 4) {
        // A fragment: 16x4 f32 -> 2 VGPRs; lane half selects K pair {0,1}/{2,3}
        v2f a;
        a.x = Arow[k + khalf] * nrm;
        a.y = Arow[k + khalf + 1] * nrm;
#pragma unroll
        for (int t = 0; t < 8; ++t) {
            // B fragment: 4x16 f32, rows striped across lanes (mirror of A)
            v2f b;
            b.x = Wlds[(k + khalf) * NFEAT + t * 16 + lrow];
            b.y = Wlds[(k + khalf + 1) * NFEAT + t * 16 + lrow];
            acc[t] = __builtin_amdgcn_wmma_f32_16x16x4_f32(
                false, a, false, b, (short)0, acc[t], false, false);
        }
    }

#pragma unroll
    for (int t = 0; t < 8; ++t) {
#pragma unroll
        for (int r = 0; r < 8; ++r) {
            float v = acc[t][r];
            if (relu) v = fmaxf(v, 0.0f);
            Y[(long)(rbase + rowhi + r) * NFEAT + t * 16 + lrow] = v;
        }
    }
}

// ---------------------------------------------------------------- sum pooling
__global__ void pool_kernel(const float* __restrict__ x, const int* __restrict__ gid,
                            float* __restrict__ g, int N) {
    long t = (long)blockIdx.x * blockDim.x + threadIdx.x;  // t = i*128 + f
    long i = t >> 7;
    int  f = (int)(t & 127);
    if (i < N) atomicAdd(&g[(long)gid[i] * NFEAT + f], x[t]);
}

// ---------------------------------------------------------------- head (128x200)
__global__ void mlp2_kernel(const float* __restrict__ hid, const float* __restrict__ W,
                            const float* __restrict__ b, float* __restrict__ out) {
    int t = blockIdx.x * blockDim.x + threadIdx.x;  // [0, 512*200)
    if (t >= NGRAPH * NREL) return;
    int gi = t / NREL, n = t % NREL;
    const float* h = hid + (long)gi * NFEAT;
    float s = b[n];
#pragma unroll 8
    for (int k = 0; k < NFEAT; ++k) s = fmaf(h[k], W[k * NREL + n], s);
    out[t] = s;
}

// ---------------------------------------------------------------- launcher
extern "C" void kernel_launch(void* const* d_in, const int* in_sizes, int n_in,
                              void* d_out, int out_size, void* d_ws, size_t ws_size,
                              hipStream_t stream) {
    const int*   nid    = (const int*)d_in[0];
    const int*   z      = (const int*)d_in[1];
    const int*   src    = (const int*)d_in[2];
    const int*   dst    = (const int*)d_in[3];
    const int*   gid    = (const int*)d_in[4];
    const float* emb    = (const float*)d_in[5];
    const float* ztab   = (const float*)d_in[6];
    const float* Ws[3]  = {(const float*)d_in[7], (const float*)d_in[9], (const float*)d_in[11]};
    const float* bs[3]  = {(const float*)d_in[8], (const float*)d_in[10], (const float*)d_in[12]};
    const float* lin1_W = (const float*)d_in[13];
    const float* lin1_b = (const float*)d_in[14];
    const float* lin2_W = (const float*)d_in[15];
    const float* lin2_b = (const float*)d_in[16];

    const int N = in_sizes[0];
    const int E = in_sizes[2];

    // workspace layout (floats)
    float* x        = (float*)d_ws;                 // N*128
    float* agg      = x + (size_t)N * NFEAT;        // N*128
    float* norm_out = agg + (size_t)N * NFEAT;      // N    (contiguous with norm_in)
    float* norm_in  = norm_out + N;                 // N
    float* g        = norm_in + N;                  // 512*128
    float* hid      = g + (size_t)NGRAPH * NFEAT;   // 512*128

    const int TB = 256;
    auto blocks = [](long n, int tb) { return (unsigned)((n + tb - 1) / tb); };

    // 1) degree norms (shared by all three layers)
    zero_f32_kernel<<<blocks(2L * N, TB), TB, 0, stream>>>(norm_out, 2L * N);
    degree_kernel<<<blocks(E, TB), TB, 0, stream>>>(src, dst, norm_out, norm_in, E);
    norm_finalize_kernel<<<blocks(2L * N, TB), TB, 0, stream>>>(norm_out, 2L * N);

    // 2) x = concat(init_embed[nid], z_table[z])
    build_x_kernel<<<blocks((long)N * NFEAT, TB), TB, 0, stream>>>(nid, z, emb, ztab, x, N);

    // 3) three GCN layers
    const unsigned gemm_blocks = (unsigned)((N + 127) / 128);
    for (int layer = 0; layer < 3; ++layer) {
        zero_f32_kernel<<<blocks((long)N * NFEAT, TB), TB, 0, stream>>>(agg, (long)N * NFEAT);
        scatter_kernel<<<blocks((long)E * 32, TB), TB, 0, stream>>>(src, dst, x, norm_out, agg, E);
        gemm128_wmma_kernel<<<gemm_blocks, TB, 0, stream>>>(
            agg, norm_in, Ws[layer], bs[layer], x, N, layer < 2 ? 1 : 0);
    }

    // 4) sum pooling over graphs
    zero_f32_kernel<<<blocks((long)NGRAPH * NFEAT, TB), TB, 0, stream>>>(g, (long)NGRAPH * NFEAT);
    pool_kernel<<<blocks((long)N * NFEAT, TB), TB, 0, stream>>>(x, gid, g, N);

    // 5) head: hid = relu(g @ lin1_W + lin1_b), out = hid @ lin2_W + lin2_b
    gemm128_wmma_kernel<<<(NGRAPH + 127) / 128, TB, 0, stream>>>(
        g, nullptr, lin1_W, lin1_b, hid, NGRAPH, 1);
    mlp2_kernel<<<blocks((long)NGRAPH * NREL, TB), TB, 0, stream>>>(hid, lin2_W, lin2_b, (float*)d_out);
}